// GIN_4423816315318
// MI455X (gfx1250) — compile-verified
//
#include <hip/hip_runtime.h>
#include <hip/hip_bf16.h>

typedef float v2f __attribute__((ext_vector_type(2)));
typedef float v8f __attribute__((ext_vector_type(8)));

static constexpr int kNodes = 50000;
static constexpr int kEdges = 800000;
static constexpr int kInCh  = 64;
static constexpr int kHid   = 128;

// ---------------------------------------------------------------------------
// init agg = x (fuses the "+ x_i" term of GIN: agg starts at x, edges add in)
// ---------------------------------------------------------------------------
__global__ void copy4_kernel(const float4* __restrict__ s, float4* __restrict__ d, int n4) {
    int i = blockIdx.x * blockDim.x + threadIdx.x;
    if (i < n4) d[i] = s[i];
}

// ---------------------------------------------------------------------------
// edge scatter: agg[dst] += feat[src], 4 features per thread, f32 atomics.
// Scatter targets (25.6 MB) are L2-resident (192 MB L2) -> L2 RMWs, no HBM
// round-trips. grid = (ceil(E/256), d/4)
// ---------------------------------------------------------------------------
__global__ void edge_scatter_kernel(const float* __restrict__ feat,
                                    const int* __restrict__ srcIdx,
                                    const int* __restrict__ dstIdx,
                                    float* __restrict__ agg, int d) {
    int e = blockIdx.x * blockDim.x + threadIdx.x;
    if (e >= kEdges) return;
    int g = blockIdx.y << 2;
    int s = srcIdx[e];
    int t = dstIdx[e];
    float4 v = *(const float4*)(feat + (size_t)s * d + g);
    float* p = agg + (size_t)t * d + g;
    atomicAdd(p + 0, v.x);
    atomicAdd(p + 1, v.y);
    atomicAdd(p + 2, v.z);
    atomicAdd(p + 3, v.w);
}

// ---------------------------------------------------------------------------
// WMMA f32 GEMM: C[50000 x 128] = act(A[50000 x K] * B[K x 128] + bias)
// One wave computes a 16-row x 128-col strip (8 accumulator tiles).
//
// B is staged in LDS pre-packed in WMMA-fragment order:
//   slot (q*128 + n)*2 + lh  (a float2)  =  ( B[4q+2*lh][n], B[4q+2*lh+1][n] )
// so each B fragment is ONE ds_load_b64 into an even-aligned VGPR pair that
// feeds v_wmma directly (no repack movs). Dword addr = q*512 + 4n + 2lh:
// lanes 0-15 use banks {4m,4m+1}, lanes 16-31 use {4m+2,4m+3} -> 64 distinct
// banks, conflict-free, no padding (K=128 case is exactly 64KB LDS).
//
// A fragment (32-bit A 16x4, ISA 7.12.2): lane L -> m=L&15, koff=2*(L>=16),
//   v0 = A[m][k+koff], v1 = A[m][k+koff+1]   (one global_load_b64)
// C/D: VGPR j -> rows j (lanes 0-15) / j+8 (lanes 16-31), col = lane&15
// ---------------------------------------------------------------------------
template <int K, bool RELU>
__global__ __launch_bounds__(256)
void gemm_wmma_kernel(const float* __restrict__ A,
                      const float* __restrict__ B,     // [K, 128] row-major
                      const float* __restrict__ bias,  // [128]
                      float* __restrict__ C) {
    __shared__ v2f Bs[(K / 4) * kHid * 2];  // == K*kHid floats: 32KB / 64KB

    // cooperative fill in packed-fragment order
    for (int i = threadIdx.x; i < (K / 4) * kHid * 2; i += blockDim.x) {
        int q    = i >> 8;        // i / (kHid*2)
        int r    = i & 255;
        int n    = r >> 1;
        int lh   = r & 1;
        int krow = 4 * q + 2 * lh;
        v2f p;
        p.x = B[(size_t)krow * kHid + n];
        p.y = B[(size_t)(krow + 1) * kHid + n];
        Bs[i] = p;
    }
    __syncthreads();

    const int wave = threadIdx.x >> 5;
    const int lane = threadIdx.x & 31;
    const int rowTile = blockIdx.x * 8 + wave;
    if (rowTile >= kNodes / 16) return;   // whole-wave uniform exit

    const int row0 = rowTile * 16;
    const int m    = lane & 15;
    const int lh   = lane >> 4;           // 0 or 1
    const int koff = lh << 1;             // 0 or 2
    const float* arow = A + (size_t)(row0 + m) * K;

    const v8f vzero = {0.f, 0.f, 0.f, 0.f, 0.f, 0.f, 0.f, 0.f};
    v8f acc[8];
#pragma unroll
    for (int t = 0; t < 8; ++t) acc[t] = vzero;

    for (int k = 0; k < K; k += 4) {
        v2f a;
        a.x = arow[k + koff];
        a.y = arow[k + koff + 1];
        const v2f* brow = Bs + ((k >> 2) * kHid) * 2 + lh;
#pragma unroll
        for (int t = 0; t < 8; ++t) {
            v2f b = brow[(t * 16 + m) * 2];
            acc[t] = __builtin_amdgcn_wmma_f32_16x16x4_f32(
                false, a, false, b, (short)0, acc[t], false, false);
        }
    }

    const int mo = lh * 8;
#pragma unroll
    for (int t = 0; t < 8; ++t) {
        const int col = t * 16 + m;
        const float bv = bias[col];
#pragma unroll
        for (int j = 0; j < 8; ++j) {
            float v = acc[t][j] + bv;
            if (RELU) v = fmaxf(v, 0.f);
            C[(size_t)(row0 + mo + j) * kHid + col] = v;
        }
    }
}

// ---------------------------------------------------------------------------
// out[i] = h[i,:] . Wout + bout   (HID=128 -> 1)
// ---------------------------------------------------------------------------
__global__ __launch_bounds__(256)
void out_proj_kernel(const float* __restrict__ H,
                     const float* __restrict__ Wout,
                     const float* __restrict__ bout,
                     float* __restrict__ out) {
    __shared__ float w[kHid];
    if (threadIdx.x < kHid) w[threadIdx.x] = Wout[threadIdx.x];
    __syncthreads();
    int i = blockIdx.x * blockDim.x + threadIdx.x;
    if (i >= kNodes) return;
    const float* h = H + (size_t)i * kHid;
    float s = 0.f;
#pragma unroll 8
    for (int k = 0; k < kHid; ++k) s = fmaf(h[k], w[k], s);
    out[i] = s + bout[0];
}

// ---------------------------------------------------------------------------
extern "C" void kernel_launch(void* const* d_in, const int* in_sizes, int n_in,
                              void* d_out, int out_size, void* d_ws, size_t ws_size,
                              hipStream_t stream) {
    const float* x    = (const float*)d_in[0];
    const int*   ei   = (const int*)d_in[1];        // [2, E] int32
    const float* W1   = (const float*)d_in[2];
    const float* b1   = (const float*)d_in[3];
    const float* W2   = (const float*)d_in[4];
    const float* b2   = (const float*)d_in[5];
    const float* W3   = (const float*)d_in[6];
    const float* b3   = (const float*)d_in[7];
    const float* W4   = (const float*)d_in[8];
    const float* b4   = (const float*)d_in[9];
    const float* Wout = (const float*)d_in[10];
    const float* bout = (const float*)d_in[11];
    float* out = (float*)d_out;

    const int* srcIdx = ei;
    const int* dstIdx = ei + kEdges;

    // workspace carve-out (total 89.6 MB of f32)
    float* ws   = (float*)d_ws;
    float* agg1 = ws;                             // [N, 64]
    float* tmp  = agg1 + (size_t)kNodes * kInCh;  // [N, 128]
    float* h1   = tmp  + (size_t)kNodes * kHid;   // [N, 128] (reused for h2)
    float* agg2 = h1   + (size_t)kNodes * kHid;   // [N, 128]

    const int edgeBlocks = (kEdges + 255) / 256;
    const int gemmBlocks = (kNodes / 16 + 7) / 8;   // 3125 tiles / 8 waves

    // ---- layer 1: agg1 = x + scatter(x) ----
    {
        int n4 = kNodes * kInCh / 4;
        copy4_kernel<<<(n4 + 255) / 256, 256, 0, stream>>>((const float4*)x, (float4*)agg1, n4);
        edge_scatter_kernel<<<dim3(edgeBlocks, kInCh / 4), 256, 0, stream>>>(
            x, srcIdx, dstIdx, agg1, kInCh);
    }
    // MLP1 + outer relu
    gemm_wmma_kernel<kInCh, true><<<gemmBlocks, 256, 0, stream>>>(agg1, W1, b1, tmp);
    gemm_wmma_kernel<kHid,  true><<<gemmBlocks, 256, 0, stream>>>(tmp,  W2, b2, h1);

    // ---- layer 2: agg2 = h1 + scatter(h1) ----
    {
        int n4 = kNodes * kHid / 4;
        copy4_kernel<<<(n4 + 255) / 256, 256, 0, stream>>>((const float4*)h1, (float4*)agg2, n4);
        edge_scatter_kernel<<<dim3(edgeBlocks, kHid / 4), 256, 0, stream>>>(
            h1, srcIdx, dstIdx, agg2, kHid);
    }
    // MLP2 + outer relu (h2 overwrites h1 buffer)
    gemm_wmma_kernel<kHid, true><<<gemmBlocks, 256, 0, stream>>>(agg2, W3, b3, tmp);
    gemm_wmma_kernel<kHid, true><<<gemmBlocks, 256, 0, stream>>>(tmp,  W4, b4, h1);

    // ---- output projection ----
    out_proj_kernel<<<(kNodes + 255) / 256, 256, 0, stream>>>(h1, Wout, bout, out);
}